// GCN_GMM_59442347377128
// MI455X (gfx1250) — compile-verified
//
#include <hip/hip_runtime.h>
#include <hip/hip_bf16.h>

typedef __attribute__((ext_vector_type(2))) float v2f;
typedef __attribute__((ext_vector_type(8))) float v8f;

#define D_FEAT 128

// ---------------------------------------------------------------------------
// 1) init: zero the output accumulator, set deg = 1.0 (self-loop contribution)
// ---------------------------------------------------------------------------
__global__ __launch_bounds__(256) void gcn_init(float* __restrict__ agg,
                                                float* __restrict__ deg,
                                                int n_nodes) {
    size_t i = (size_t)blockIdx.x * blockDim.x + threadIdx.x;
    size_t total = (size_t)n_nodes * D_FEAT;
    if (i < total) agg[i] = 0.0f;
    if (i < (size_t)n_nodes) deg[i] = 1.0f;
}

// ---------------------------------------------------------------------------
// 2) degree: deg[dst] += 1 for every real edge (self-loops pre-counted)
// ---------------------------------------------------------------------------
__global__ __launch_bounds__(256) void gcn_degree(const int* __restrict__ ei,
                                                  float* __restrict__ deg,
                                                  int n_edges) {
    int e = blockIdx.x * blockDim.x + threadIdx.x;
    if (e < n_edges) {
        int d = ei[n_edges + e];   // edge_index[1][e]
        atomicAdd(&deg[d], 1.0f);
    }
}

// ---------------------------------------------------------------------------
// 3) dinv = rsqrt(deg) in place (deg >= 1 always, no zero guard needed)
// ---------------------------------------------------------------------------
__global__ __launch_bounds__(256) void gcn_rsqrt(float* __restrict__ deg, int n_nodes) {
    int i = blockIdx.x * blockDim.x + threadIdx.x;
    if (i < n_nodes) deg[i] = rsqrtf(deg[i]);
}

// ---------------------------------------------------------------------------
// 4) h = x @ W via V_WMMA_F32_16X16X4_F32.
//    One workgroup (256 thr = 8 waves) computes a 16-row strip; wave w owns
//    the 16x16 tile at columns [16w, 16w+16). K-loop: 32 x wmma 16x16x4.
//    x tile (16x128, 8KB) staged in LDS with coalesced loads; W is 64KB and
//    stays hot in L2 (192MB), read directly.
// ---------------------------------------------------------------------------
__global__ __launch_bounds__(256) void gcn_gemm_wmma(const float* __restrict__ x,
                                                     const float* __restrict__ W,
                                                     float* __restrict__ h,
                                                     int n_nodes) {
    __shared__ float xs[16 * D_FEAT];   // 8 KB

    const int tid  = threadIdx.x;
    const int wave = tid >> 5;
    const int lane = tid & 31;
    const int rowBase = blockIdx.x * 16;

    // Cooperative, coalesced load of the 16x128 x tile.
    for (int i = tid; i < 16 * D_FEAT; i += 256) {
        int r = i >> 7;           // row in tile
        int c = i & (D_FEAT - 1); // column
        int grow = rowBase + r;
        xs[i] = (grow < n_nodes) ? x[(size_t)grow * D_FEAT + c] : 0.0f;
    }
    __syncthreads();

    const int colBase = wave * 16;
    const int mrow  = lane & 15;          // matrix row (A) / column (B,C)
    const int khalf = (lane >> 4) << 1;   // lanes 0-15 -> K+0,K+1 ; lanes 16-31 -> K+2,K+3

    v8f c = {};
    for (int k = 0; k < D_FEAT; k += 4) {
        // A fragment: 16x4 f32. Lane holds A[mrow][k+khalf], A[mrow][k+khalf+1]
        v2f a;
        a.x = xs[mrow * D_FEAT + k + khalf];
        a.y = xs[mrow * D_FEAT + k + khalf + 1];
        // B fragment: 4x16 f32, row-striped across lanes (half-wave per K pair).
        v2f bfrag;
        bfrag.x = W[(size_t)(k + khalf)     * D_FEAT + colBase + mrow];
        bfrag.y = W[(size_t)(k + khalf + 1) * D_FEAT + colBase + mrow];
        // D = A*B + C   (8 args: neg_a, A, neg_b, B, c_mod, C, reuse_a, reuse_b)
        c = __builtin_amdgcn_wmma_f32_16x16x4_f32(false, a, false, bfrag,
                                                  (short)0, c, false, false);
    }

    // C/D layout: VGPR v -> row M = v + 8*(lane>=16), col N = lane&15.
    const int rOff = (lane >> 4) << 3;
#pragma unroll
    for (int v = 0; v < 8; ++v) {
        int row = rowBase + v + rOff;
        if (row < n_nodes) h[(size_t)row * D_FEAT + colBase + mrow] = c[v];
    }
}

// ---------------------------------------------------------------------------
// 5) scatter: one wave32 per edge. agg[dst] += h[src] * dinv[src]*dinv[dst].
//    Each lane handles 4 floats 32 apart -> coalesced 128B per step, 512B per edge;
//    atomics resolve in L2 (agg = 51MB << 192MB L2).
// ---------------------------------------------------------------------------
__global__ __launch_bounds__(256) void gcn_scatter(const int* __restrict__ ei,
                                                   const float* __restrict__ dinv,
                                                   const float* __restrict__ h,
                                                   float* __restrict__ agg,
                                                   int n_edges) {
    int warp = (int)(((size_t)blockIdx.x * blockDim.x + threadIdx.x) >> 5);
    int lane = threadIdx.x & 31;
    if (warp >= n_edges) return;

    int s = ei[warp];            // edge_index[0][e]
    int d = ei[n_edges + warp];  // edge_index[1][e]
    float w = dinv[s] * dinv[d];

    const float* hs = h   + (size_t)s * D_FEAT;
    float*       ad = agg + (size_t)d * D_FEAT;
#pragma unroll
    for (int j = 0; j < 4; ++j) {
        int f = lane + j * 32;
        atomicAdd(&ad[f], hs[f] * w);
    }
}

// ---------------------------------------------------------------------------
// 6) finalize: out = relu(agg + h*dinv^2 (self-loop) + b)
// ---------------------------------------------------------------------------
__global__ __launch_bounds__(256) void gcn_finalize(float* __restrict__ out,
                                                    const float* __restrict__ h,
                                                    const float* __restrict__ dinv,
                                                    const float* __restrict__ b,
                                                    int n_nodes) {
    size_t i = (size_t)blockIdx.x * blockDim.x + threadIdx.x;
    if (i >= (size_t)n_nodes * D_FEAT) return;
    int node = (int)(i >> 7);
    int f    = (int)(i & (D_FEAT - 1));
    float di = dinv[node];
    float v  = out[i] + h[i] * di * di + b[f];
    out[i] = fmaxf(v, 0.0f);
}

// ---------------------------------------------------------------------------
extern "C" void kernel_launch(void* const* d_in, const int* in_sizes, int n_in,
                              void* d_out, int out_size, void* d_ws, size_t ws_size,
                              hipStream_t stream) {
    const float* x  = (const float*)d_in[0];
    const int*   ei = (const int*)d_in[1];   // edge_index, [2, E] row-major (int32)
    const float* W  = (const float*)d_in[2];
    const float* b  = (const float*)d_in[3];
    float* out = (float*)d_out;

    const int n_nodes = in_sizes[0] / D_FEAT;   // 100000
    const int n_edges = in_sizes[1] / 2;        // 3200000

    // Workspace layout: [deg/dinv : n_nodes floats][pad][h : n_nodes*128 floats]
    float* deg = (float*)d_ws;
    size_t degSlots = ((size_t)n_nodes + 1023) & ~(size_t)1023;
    float* h = deg + degSlots;

    const size_t total = (size_t)n_nodes * D_FEAT;
    const int blk = 256;

    // 1) zero accumulator, deg = 1 (self-loops)
    gcn_init<<<(unsigned)((total + blk - 1) / blk), blk, 0, stream>>>(out, deg, n_nodes);

    // 2) degree count over real edges
    gcn_degree<<<(n_edges + blk - 1) / blk, blk, 0, stream>>>(ei, deg, n_edges);

    // 3) dinv = rsqrt(deg)
    gcn_rsqrt<<<(n_nodes + blk - 1) / blk, blk, 0, stream>>>(deg, n_nodes);

    // 4) h = x @ W  (WMMA f32 16x16x4)
    gcn_gemm_wmma<<<(n_nodes + 15) / 16, blk, 0, stream>>>(x, W, h, n_nodes);

    // 5) edge scatter: one wave per edge
    {
        size_t warps = (size_t)n_edges;
        unsigned grid = (unsigned)((warps * 32 + blk - 1) / blk);
        gcn_scatter<<<grid, blk, 0, stream>>>(ei, deg, h, out, n_edges);
    }

    // 6) bias + self-loop + ReLU
    gcn_finalize<<<(unsigned)((total + blk - 1) / blk), blk, 0, stream>>>(out, h, deg, b, n_nodes);
}